// LSTMModel_83691732729946
// MI455X (gfx1250) — compile-verified
//
#include <hip/hip_runtime.h>

// ---------------------------------------------------------------------------
// LSTM classifier (B=4096, T=200, IN=34, H=4, C=3) for MI455X / gfx1250.
//
// Roofline: bytes dominate (x: 112MB read, x_gates: 52MB write + 52MB read
// => ~217MB => ~9.3us @ 23.3TB/s). FLOPs ~1G, negligible. Strategy:
//   K1: input projection as fp32 WMMA (V_WMMA_F32_16X16X4_F32) -> x_gates[ws]
//   K2: sequential scan, 4 lanes per batch element, native v_tanh/v_exp,
//       global_prefetch for the streaming x_gates reads, fused final FC.
// ---------------------------------------------------------------------------

#define BATCH 4096
#define TLEN  200
#define INSZ  34
#define HID   4
#define GATES 16   // 4*HID
#define NCLS  3
#define MROWS (BATCH * TLEN)        // 819200, divisible by 16

typedef float v2f __attribute__((ext_vector_type(2)));
typedef float v8f __attribute__((ext_vector_type(8)));

__device__ __forceinline__ float fast_rcp(float x) {
  return __builtin_amdgcn_rcpf(x);
}
__device__ __forceinline__ float fast_sigmoid(float x) {
  return fast_rcp(1.0f + __expf(-x));           // v_exp_f32 + v_rcp_f32
}
__device__ __forceinline__ float fast_tanh(float x) {
#if __has_builtin(__builtin_amdgcn_tanhf)
  return __builtin_amdgcn_tanhf(x);             // gfx1250 v_tanh_f32
#else
  float e = __expf(-2.0f * x);
  return (1.0f - e) * fast_rcp(1.0f + e);
#endif
}

// ---------------------------------------------------------------------------
// Kernel 1: x_gates[m][g] = sum_k x[m][k] * W_ih[g][k] + b_ih[g] + b_hh[g]
//   m = b*T + t (x is [B][T][34] row-major, so rows are contiguous in m).
//   One wave = one 16x16 tile. K=34 padded to 36 -> 9 x WMMA f32 16x16x4.
//   EXEC stays all-ones (OOB-K handled with selects, no branches) per the
//   WMMA restriction in ISA 7.12.
//
// Fragment layouts (ISA 7.12.2):
//   A (16x4 f32): lanes 0-15 M=0..15 {K0,K1}; lanes 16-31 M=0..15 {K2,K3}.
//   B (4x16 f32): lanes 0-15 N=0..15 {K0,K1}; lanes 16-31 N=0..15 {K2,K3}.
//   C/D (16x16):  VGPR v: lanes 0-15 -> M=v, lanes 16-31 -> M=v+8; N=lane&15.
// ---------------------------------------------------------------------------
__global__ __launch_bounds__(256) void lstm_xgates_wmma(
    const float* __restrict__ x,     // [MROWS][34]
    const float* __restrict__ W_ih,  // [16][34]
    const float* __restrict__ b_ih,  // [16]
    const float* __restrict__ b_hh,  // [16]
    float* __restrict__ xg)          // [MROWS][16]
{
  const int lane = threadIdx.x & 31;
  const int wave = threadIdx.x >> 5;
  const int half = lane >> 4;              // 0: lanes 0-15, 1: lanes 16-31
  const int l    = lane & 15;
  const int koff = half * 2;               // K offset for upper half-wave
  const long long m0 = ((long long)blockIdx.x * 8 + wave) * 16;

  // Accumulator preloaded with the (column-only) bias: C[m][n] = b_ih[n]+b_hh[n]
  const float bias = b_ih[l] + b_hh[l];
  v8f acc;
#pragma unroll
  for (int v = 0; v < 8; ++v) acc[v] = bias;

  const float* __restrict__ xrow = x    + (m0 + l) * INSZ;  // A row for this lane
  const float* __restrict__ wrow = W_ih + l * INSZ;         // B col (= W_ih row) for this lane

#pragma unroll
  for (int kk = 0; kk < 9; ++kk) {
    const int k0 = kk * 4 + koff;
    // Clamp address, zero via select -> no EXEC divergence around WMMA.
    float a0 = xrow[min(k0,     INSZ - 1)];
    float a1 = xrow[min(k0 + 1, INSZ - 1)];
    float w0 = wrow[min(k0,     INSZ - 1)];
    float w1 = wrow[min(k0 + 1, INSZ - 1)];
    v2f a, b;
    a.x = (k0     < INSZ) ? a0 : 0.0f;
    a.y = (k0 + 1 < INSZ) ? a1 : 0.0f;
    b.x = (k0     < INSZ) ? w0 : 0.0f;
    b.y = (k0 + 1 < INSZ) ? w1 : 0.0f;
    // D = A(16x4) x B(4x16) + C, fp32 throughout
    acc = __builtin_amdgcn_wmma_f32_16x16x4_f32(
        false, a, false, b, (short)0, acc, false, false);
  }

  // Scatter tile: row m = m0 + v + 8*half, col = l. Tile spans one contiguous
  // 1KB block of xg -> coalesces well.
#pragma unroll
  for (int v = 0; v < 8; ++v) {
    const long long m = m0 + v + 8 * half;
    xg[m * GATES + l] = acc[v];
  }
}

// ---------------------------------------------------------------------------
// Kernel 2: sequential LSTM scan + fused FC.
//   4 lanes per batch element; lane j owns h_j, c_j and gate rows
//   {j, 4+j, 8+j, 12+j} of W_hh. Per (b,t) the 4-lane group consumes exactly
//   one 64B line of x_gates. h broadcast via __shfl width=4.
//   16384 threads = 512 waves -> decent latency hiding for the T-serial loop,
//   helped by global_prefetch_b8 8 steps ahead.
// ---------------------------------------------------------------------------
__global__ __launch_bounds__(256) void lstm_scan(
    const float* __restrict__ xg,    // [B][T][16]
    const float* __restrict__ W_hh,  // [16][4]
    const float* __restrict__ W_fc,  // [3][4]
    const float* __restrict__ b_fc,  // [3]
    float* __restrict__ out)         // [B][3]
{
  const int gtid = blockIdx.x * blockDim.x + threadIdx.x;
  const int b = gtid >> 2;
  const int j = gtid & 3;
  if (b >= BATCH) return;

  // Recurrent weight rows owned by this lane (16B-aligned float4 loads).
  const float4 Wi = *(const float4*)(W_hh + (j     ) * HID);
  const float4 Wf = *(const float4*)(W_hh + (4  + j) * HID);
  const float4 Wg = *(const float4*)(W_hh + (8  + j) * HID);
  const float4 Wo = *(const float4*)(W_hh + (12 + j) * HID);

  const float* __restrict__ xgb = xg + (long long)b * TLEN * GATES;

  float h = 0.0f, c = 0.0f;
  for (int t = 0; t < TLEN; ++t) {
    const float* __restrict__ p = xgb + t * GATES;
    __builtin_prefetch(p + 8 * GATES, 0, 0);   // global_prefetch_b8, ~8 steps ahead

    float pi = p[j];
    float pf = p[4  + j];
    float pg = p[8  + j];
    float po = p[12 + j];

    // Broadcast h[0..3] within the 4-lane group.
    const float h0 = __shfl(h, 0, 4);
    const float h1 = __shfl(h, 1, 4);
    const float h2 = __shfl(h, 2, 4);
    const float h3 = __shfl(h, 3, 4);

    pi += Wi.x * h0 + Wi.y * h1 + Wi.z * h2 + Wi.w * h3;
    pf += Wf.x * h0 + Wf.y * h1 + Wf.z * h2 + Wf.w * h3;
    pg += Wg.x * h0 + Wg.y * h1 + Wg.z * h2 + Wg.w * h3;
    po += Wo.x * h0 + Wo.y * h1 + Wo.z * h2 + Wo.w * h3;

    const float ig = fast_sigmoid(pi);
    const float fg = fast_sigmoid(pf);
    const float gg = fast_tanh(pg);
    const float og = fast_sigmoid(po);

    c = fg * c + ig * gg;
    h = og * fast_tanh(c);
  }

  // Fused FC: out[b][n] = b_fc[n] + sum_k h_k * W_fc[n][k], n = j for j < 3.
  const float h0 = __shfl(h, 0, 4);
  const float h1 = __shfl(h, 1, 4);
  const float h2 = __shfl(h, 2, 4);
  const float h3 = __shfl(h, 3, 4);
  if (j < NCLS) {
    const float4 Wc = *(const float4*)(W_fc + j * HID);
    out[b * NCLS + j] =
        b_fc[j] + Wc.x * h0 + Wc.y * h1 + Wc.z * h2 + Wc.w * h3;
  }
}

// ---------------------------------------------------------------------------
extern "C" void kernel_launch(void* const* d_in, const int* in_sizes, int n_in,
                              void* d_out, int out_size, void* d_ws, size_t ws_size,
                              hipStream_t stream) {
  (void)in_sizes; (void)n_in; (void)out_size; (void)ws_size;
  const float* x    = (const float*)d_in[0];  // [4096,200,34]
  const float* W_ih = (const float*)d_in[1];  // [16,34]
  const float* W_hh = (const float*)d_in[2];  // [16,4]
  const float* b_ih = (const float*)d_in[3];  // [16]
  const float* b_hh = (const float*)d_in[4];  // [16]
  const float* W_fc = (const float*)d_in[5];  // [3,4]
  const float* b_fc = (const float*)d_in[6];  // [3]
  float* out = (float*)d_out;                 // [4096,3]
  float* xg  = (float*)d_ws;                  // 819200*16 floats = 52.4 MB scratch

  // K1: 51200 tiles of 16 rows, 8 waves (tiles) per 256-thread block.
  const int tiles  = MROWS / 16;              // 51200
  const int blocks1 = tiles / 8;              // 6400
  hipLaunchKernelGGL(lstm_xgates_wmma, dim3(blocks1), dim3(256), 0, stream,
                     x, W_ih, b_ih, b_hh, xg);

  // K2: 4 lanes per batch element -> 16384 threads = 64 blocks.
  const int blocks2 = (BATCH * 4) / 256;      // 64
  hipLaunchKernelGGL(lstm_scan, dim3(blocks2), dim3(256), 0, stream,
                     xg, W_hh, W_fc, b_fc, out);
}